// SwitchFocusedLossNoSegmentation_80221399155359
// MI455X (gfx1250) — compile-verified
//
#include <hip/hip_runtime.h>
#include <cstdint>

#define S_LEN 8192
#define TOK_PER_THREAD 32
#define BLOCK 256

// OR over bits [j-5, j+5] of the 96-bit window lo:mid:hi, result bit j.
__device__ __forceinline__ unsigned dilate5(unsigned lo, unsigned mid, unsigned hi) {
  unsigned r = mid;
#pragma unroll
  for (int d = 1; d <= 5; ++d)
    r |= (mid >> d) | (hi << (32 - d)) | (mid << d) | (lo >> (32 - d));
  return r;
}

__global__ __launch_bounds__(BLOCK) void row_loss_kernel(
    const float* __restrict__ logits, const uint64_t labels_base,
    float* __restrict__ ws) {
  __shared__ int      labLds[S_LEN];     // staged low-dwords of int64 labels (32 KB)
  __shared__ unsigned predB[S_LEN / 32]; // predicted-switch bitmask, 1 bit/token
  __shared__ unsigned trueB[S_LEN / 32]; // true-switch bitmask
  __shared__ float    redY[8], redN[8];
  __shared__ int      redC[8];
  __shared__ unsigned hasT;

  const int tid  = threadIdx.x;
  const int row  = blockIdx.x;
  const int lane = tid & 31;
  const int wv   = tid >> 5;
  if (tid == 0) hasT = 0u;

  const uint64_t rowTok = (uint64_t)row * S_LEN;

  // --- CDNA5 async-to-LDS staging of this thread's 32 label dwords ---------
  // token = k*256 + tid; label int64 at byte (rowTok+token)*8, low dword only.
#pragma unroll
  for (int k = 0; k < TOK_PER_THREAD; ++k) {
    const int token  = k * BLOCK + tid;
    uint32_t ldsOff  = (uint32_t)(uintptr_t)(&labLds[token]); // low 32 bits = LDS byte offset
    uint64_t gaddr   = labels_base + ((rowTok + (uint64_t)token) << 3);
    asm volatile("global_load_async_to_lds_b32 %0, %1, off"
                 :: "v"(ldsOff), "v"(gaddr) : "memory");
  }
  asm volatile("s_wait_asynccnt 0x0" ::: "memory"); // this wave's copies done

  const float4* lg4 = (const float4*)logits + rowTok;

  // --- Phase 1: per-token loss + ballot-built switch masks -----------------
  float a[TOK_PER_THREAD];
  int vcnt = 0;
#pragma unroll
  for (int k = 0; k < TOK_PER_THREAD; ++k) {
    const int token = k * BLOCK + tid;
    float4 lg = lg4[token];                 // coalesced b128 stream
    int lab = labLds[token];
    bool valid = (lab != -100);
    int slab = valid ? lab : 0;
    float m01 = fmaxf(lg.x, lg.y);
    float m23 = fmaxf(lg.z, lg.w);
    float m   = fmaxf(m01, m23);
    float es  = __expf(lg.x - m) + __expf(lg.y - m) + __expf(lg.z - m) + __expf(lg.w - m);
    float lse = m + __logf(es);
    float sel = (slab == 0) ? lg.x : (slab == 1) ? lg.y : (slab == 2) ? lg.z : lg.w;
    float wgt = (slab >= 2) ? 5.0f : 0.1f;
    a[k] = valid ? (lse - sel) * wgt : 0.0f;
    vcnt += valid ? 1 : 0;
    // argmax>=2  <=>  max(l2,l3) strictly beats max(l0,l1) (argmax ties -> lower idx)
    unsigned pb = __builtin_amdgcn_ballot_w32(m23 > m01);
    unsigned tb = __builtin_amdgcn_ballot_w32(lab >= 2); // -100 -> false
    if (lane == 0) {                 // token = (k*8+wv)*32 + lane  => exact word
      const int wd = k * 8 + wv;
      predB[wd] = pb;
      trueB[wd] = tb;
    }
  }
  __syncthreads();

  if (trueB[tid] != 0u) atomicOr(&hasT, 1u);
  __syncthreads();
  const bool rowHasTrue = (hasT != 0u);

  // --- Phase 2: dilate masks, apply factors, accumulate both hypotheses ----
  float sy = 0.0f, sn = 0.0f;
#pragma unroll
  for (int k = 0; k < TOK_PER_THREAD; ++k) {
    const int wd = k * 8 + wv;                    // word containing this token
    unsigned pm = predB[wd];
    unsigned tm = trueB[wd];
    unsigned pl = (wd > 0)   ? predB[wd - 1] : 0u;
    unsigned ph = (wd < 255) ? predB[wd + 1] : 0u;
    unsigned tl = (wd > 0)   ? trueB[wd - 1] : 0u;
    unsigned th = (wd < 255) ? trueB[wd + 1] : 0u;
    unsigned pn = dilate5(pl, pm, ph);
    unsigned tn = dilate5(tl, tm, th);
    float A = a[k];
    bool tsw = (tm >> lane) & 1u;
    bool psw = (pm >> lane) & 1u;
    bool pnb = (pn >> lane) & 1u;
    bool tnb = (tn >> lane) & 1u;
    // reward and penalty are mutually exclusive (tsw => tnb)
    float f = (tsw && pnb) ? 0.1f : ((psw && !tnb) ? 2.0f : 1.0f);
    sy += A * f;   // value if row has a true switch
    sn += A;       // value if row has none (no factor can fire)
  }

  // --- Deterministic block reduction ---------------------------------------
#pragma unroll
  for (int off = 16; off > 0; off >>= 1) {
    sy   += __shfl_xor(sy, off, 32);
    sn   += __shfl_xor(sn, off, 32);
    vcnt += __shfl_xor(vcnt, off, 32);
  }
  if (lane == 0) { redY[wv] = sy; redN[wv] = sn; redC[wv] = vcnt; }
  __syncthreads();
  if (tid == 0) {
    float Y = 0.f, N = 0.f; int Ct = 0;
    for (int i = 0; i < 8; ++i) { Y += redY[i]; N += redN[i]; Ct += redC[i]; }
    ws[row * 2 + 0] = rowHasTrue ? Y : N;
    ws[row * 2 + 1] = (float)Ct;
  }
}

__global__ __launch_bounds__(256) void finalize_kernel(const float* __restrict__ ws,
                                                       float* __restrict__ out, int B) {
  __shared__ double rl[256], rc[256];
  const int t = threadIdx.x;
  double L = 0.0, Cn = 0.0;
  for (int r = t; r < B; r += 256) {
    L  += (double)ws[r * 2 + 0];
    Cn += (double)ws[r * 2 + 1];
  }
  rl[t] = L; rc[t] = Cn;
  __syncthreads();
  for (int s2 = 128; s2 > 0; s2 >>= 1) {
    if (t < s2) { rl[t] += rl[t + s2]; rc[t] += rc[t + s2]; }
    __syncthreads();
  }
  if (t == 0) out[0] = (float)(rl[0] / rc[0]);
}

extern "C" void kernel_launch(void* const* d_in, const int* in_sizes, int n_in,
                              void* d_out, int out_size, void* d_ws, size_t ws_size,
                              hipStream_t stream) {
  const float*  logits = (const float*)d_in[0];
  const uint64_t labels = (uint64_t)(uintptr_t)d_in[1]; // int64 labels, raw address
  float* ws = (float*)d_ws;                             // B*2 floats (8 KB)
  const int BS = in_sizes[1];       // B*S
  const int B  = BS / S_LEN;        // S fixed at 8192, C at 4 per reference

  row_loss_kernel<<<B, BLOCK, 0, stream>>>(logits, labels, ws);
  finalize_kernel<<<1, 256, 0, stream>>>(ws, (float*)d_out, B);
}